// FlexibleGNN_24867860644044
// MI455X (gfx1250) — compile-verified
//
#include <hip/hip_runtime.h>
#include <math.h>

#define DIM 96
#define LN_EPS 1e-5f

typedef __attribute__((ext_vector_type(2))) float v2f;
typedef __attribute__((ext_vector_type(8))) float v8f;

// ---------------------------------------------------------------------------
// Degree / normalization kernels
// ---------------------------------------------------------------------------
__global__ void deg_init_kernel(float* __restrict__ deg, int n) {
    int i = blockIdx.x * blockDim.x + threadIdx.x;
    if (i < n) deg[i] = 1.0f;   // self-loop contributes 1
}

__global__ void deg_accum_kernel(const int* __restrict__ dst,
                                 float* __restrict__ deg, int E) {
    int e = blockIdx.x * blockDim.x + threadIdx.x;
    if (e < E) atomicAdd(&deg[dst[e]], 1.0f);
}

__global__ void dis_kernel(const float* __restrict__ deg,
                           float* __restrict__ dis, int n) {
    int i = blockIdx.x * blockDim.x + threadIdx.x;
    if (i < n) dis[i] = rsqrtf(deg[i]);
}

// ---------------------------------------------------------------------------
// GEMM: out[n x 96] = X[n x 96] @ W[96 x 96], fp32 WMMA 16x16x4
// One wave per 16-row block; W cached in LDS; A fragments held in 48 VGPRs.
// ---------------------------------------------------------------------------
__global__ __launch_bounds__(256)
void gemm_wmma_kernel(const float* __restrict__ X, const float* __restrict__ W,
                      float* __restrict__ out, int n, int nRowBlocks) {
    __shared__ float ldsW[DIM * DIM];

    const int tid = threadIdx.x;
    for (int i = tid; i < DIM * DIM; i += 256) ldsW[i] = W[i];
    __syncthreads();

    const int wave = tid >> 5;
    const int lane = tid & 31;
    const int m    = lane & 15;   // row within tile (A) / col within tile (B,C,D)
    const int kh   = lane >> 4;   // which K-half of the 16x4 fragment

    const int rb = blockIdx.x * 8 + wave;       // wave-uniform
    if (rb >= nRowBlocks) return;               // uniform branch: EXEC stays all-1
    const int rowBase = rb * 16;

    // Load A fragments for the whole K=96 sweep: 24 x v2f = 48 VGPRs/lane.
    int arow = rowBase + m;
    if (arow >= n) arow = n - 1;                // clamp (keeps EXEC full)
    const float* xrow = X + (size_t)arow * DIM;
    v2f a[24];
#pragma unroll
    for (int kk = 0; kk < 24; ++kk) {
        const float* p = xrow + kk * 4 + 2 * kh;
        a[kk].x = p[0];
        a[kk].y = p[1];
    }

#pragma unroll
    for (int c = 0; c < 6; ++c) {               // 6 column tiles of 16
        v8f acc = {};
#pragma unroll
        for (int kk = 0; kk < 24; ++kk) {
            const int krow = kk * 4 + 2 * kh;
            v2f b;
            b.x = ldsW[(krow + 0) * DIM + c * 16 + m];
            b.y = ldsW[(krow + 1) * DIM + c * 16 + m];
            acc = __builtin_amdgcn_wmma_f32_16x16x4_f32(
                /*neg_a=*/false, a[kk], /*neg_b=*/false, b,
                /*c_mod=*/(short)0, acc, /*reuse_a=*/false, /*reuse_b=*/false);
        }
        // C/D layout: VGPR r -> row rowBase + r + 8*kh, col c*16 + m
#pragma unroll
        for (int r = 0; r < 8; ++r) {
            const int row = rowBase + r + 8 * kh;
            if (row < n) out[(size_t)row * DIM + c * 16 + m] = acc[r];
        }
    }
}

// ---------------------------------------------------------------------------
// agg[i][:] = dis[i]^2 * hW[i][:]   (self-loop term, also initializes agg)
// ---------------------------------------------------------------------------
__global__ void self_loop_kernel(const float* __restrict__ hW,
                                 const float* __restrict__ dis,
                                 float* __restrict__ agg, int n) {
    int idx = blockIdx.x * blockDim.x + threadIdx.x;
    if (idx < n * DIM) {
        int i = idx / DIM;
        float s = dis[i];
        agg[idx] = hW[idx] * (s * s);
    }
}

// ---------------------------------------------------------------------------
// Edge scatter: agg[dst] += dis[src]*dis[dst] * hW[src]; one wave per edge.
// ---------------------------------------------------------------------------
__global__ __launch_bounds__(256)
void scatter_kernel(const int* __restrict__ src, const int* __restrict__ dst,
                    const float* __restrict__ dis, const float* __restrict__ hW,
                    float* __restrict__ agg, int E) {
    const int lane   = threadIdx.x & 31;
    const int gwave  = (blockIdx.x * blockDim.x + threadIdx.x) >> 5;
    const int nwaves = (gridDim.x * blockDim.x) >> 5;

    for (int e = gwave; e < E; e += nwaves) {
        const int s = __builtin_amdgcn_readfirstlane(src[e]);
        const int d = __builtin_amdgcn_readfirstlane(dst[e]);
        const float nrm = dis[s] * dis[d];
        const float* hrow = hW + (size_t)s * DIM;
        float*       arow = agg + (size_t)d * DIM;
#pragma unroll
        for (int j = 0; j < 3; ++j) {
            const int c = lane + 32 * j;
            atomicAdd(&arow[c], hrow[c] * nrm);
        }
    }
}

// ---------------------------------------------------------------------------
// Epilogue: out = LayerNorm(tanh(agg + b)) * g + be ; one wave per node.
// ---------------------------------------------------------------------------
__global__ __launch_bounds__(256)
void epilogue_kernel(const float* __restrict__ agg, const float* __restrict__ b,
                     const float* __restrict__ g, const float* __restrict__ be,
                     float* __restrict__ out, int n) {
    const int lane = threadIdx.x & 31;
    const int node = (blockIdx.x * blockDim.x + threadIdx.x) >> 5;
    if (node >= n) return;

    const float* arow = agg + (size_t)node * DIM;
    float t[3];
    float sum = 0.0f;
#pragma unroll
    for (int j = 0; j < 3; ++j) {
        const int c = lane + 32 * j;
        t[j] = tanhf(arow[c] + b[c]);
        sum += t[j];
    }
#pragma unroll
    for (int off = 16; off > 0; off >>= 1) sum += __shfl_xor(sum, off, 32);
    const float mu = sum * (1.0f / (float)DIM);

    float sq = 0.0f;
#pragma unroll
    for (int j = 0; j < 3; ++j) {
        t[j] -= mu;
        sq += t[j] * t[j];
    }
#pragma unroll
    for (int off = 16; off > 0; off >>= 1) sq += __shfl_xor(sq, off, 32);
    const float rs = rsqrtf(sq * (1.0f / (float)DIM) + LN_EPS);

#pragma unroll
    for (int j = 0; j < 3; ++j) {
        const int c = lane + 32 * j;
        out[(size_t)node * DIM + c] = t[j] * rs * g[c] + be[c];
    }
}

// ---------------------------------------------------------------------------
// Host launcher
// ---------------------------------------------------------------------------
static void run_layer(const float* in, const float* W, const float* b,
                      const float* g, const float* be,
                      const int* src, const int* dst, const float* dis,
                      float* hW, float* agg, float* out,
                      int n, int E, hipStream_t stream) {
    const int nRB = (n + 15) / 16;
    gemm_wmma_kernel<<<(nRB + 7) / 8, 256, 0, stream>>>(in, W, hW, n, nRB);
    self_loop_kernel<<<(n * DIM + 255) / 256, 256, 0, stream>>>(hW, dis, agg, n);
    scatter_kernel<<<2048, 256, 0, stream>>>(src, dst, dis, hW, agg, E);
    epilogue_kernel<<<(n + 7) / 8, 256, 0, stream>>>(agg, b, g, be, out, n);
}

extern "C" void kernel_launch(void* const* d_in, const int* in_sizes, int n_in,
                              void* d_out, int out_size, void* d_ws, size_t ws_size,
                              hipStream_t stream) {
    (void)n_in; (void)out_size; (void)ws_size;

    const float* x    = (const float*)d_in[0];
    const int*   edge = (const int*)d_in[1];
    const float* W0   = (const float*)d_in[2];
    const float* b0   = (const float*)d_in[3];
    const float* g0   = (const float*)d_in[4];
    const float* be0  = (const float*)d_in[5];
    const float* W1   = (const float*)d_in[6];
    const float* b1   = (const float*)d_in[7];
    const float* g1   = (const float*)d_in[8];
    const float* be1  = (const float*)d_in[9];

    const int n = in_sizes[0] / DIM;   // 50000
    const int E = in_sizes[1] / 2;     // 800000
    const int* src = edge;
    const int* dst = edge + E;

    float* ws  = (float*)d_ws;
    float* hW  = ws;                               // n*DIM
    float* agg = hW  + (size_t)n * DIM;            // n*DIM
    float* h1  = agg + (size_t)n * DIM;            // n*DIM
    float* deg = h1  + (size_t)n * DIM;            // n
    float* dis = deg + n;                          // n

    // Symmetric normalization: deg includes self-loop; dis = deg^-1/2
    deg_init_kernel<<<(n + 255) / 256, 256, 0, stream>>>(deg, n);
    deg_accum_kernel<<<(E + 255) / 256, 256, 0, stream>>>(dst, deg, E);
    dis_kernel<<<(n + 255) / 256, 256, 0, stream>>>(deg, dis, n);

    // Layer 0: x -> h1
    run_layer(x, W0, b0, g0, be0, src, dst, dis, hW, agg, h1, n, E, stream);
    // Layer 1: h1 -> out
    run_layer(h1, W1, b1, g1, be1, src, dst, dis, hW, agg, (float*)d_out, n, E, stream);
}